// GraphEmbedding_73581379715153
// MI455X (gfx1250) — compile-verified
//
#include <hip/hip_runtime.h>
#include <hip/hip_bf16.h>
#include <stdint.h>

#define BLOCK 256
#define CH    64      // rows per TDM chunk (64 rows x 64 f32 = 16 KB)
#define NBUF  3       // TDM pipeline depth (up to 2 chunks in flight + 1 computing)
#define D     64      // features per edge

typedef __attribute__((ext_vector_type(4))) unsigned int u32x4;
typedef __attribute__((ext_vector_type(8))) int          i32x8;
typedef __attribute__((ext_vector_type(4))) int          i32x4;

#if defined(__has_builtin)
#  if __has_builtin(__builtin_amdgcn_tensor_load_to_lds) && \
      __has_builtin(__builtin_amdgcn_s_wait_tensorcnt)
#    define HAVE_TDM 1
#  endif
#endif
#ifndef HAVE_TDM
#  define HAVE_TDM 0
#endif

#if HAVE_TDM
// 2D tile load: `rows` rows of 64 f32, contiguous (stride 64 elems), into LDS.
__device__ __forceinline__ void tdm_load_rows(uint32_t lds_byte_off,
                                              const float* gsrc, int rows) {
  uint64_t ga = (uint64_t)(uintptr_t)gsrc;
  u32x4 g0;
  g0[0] = 1u;                                   // count=1 (valid), user mode, no gather
  g0[1] = lds_byte_off;                         // lds_addr (bytes, wg-relative)
  g0[2] = (uint32_t)ga;                         // global_addr[31:0]
  g0[3] = ((uint32_t)(ga >> 32) & 0x01FFFFFFu)  // global_addr[56:32]
        | (2u << 30);                           // type = 2 ("image")
  uint32_t ur = (uint32_t)rows;
  i32x8 g1;
  g1[0] = (int)(2u << 16);                      // wg_mask=0, data_size=2 (4B)
  g1[1] = (int)((uint32_t)D << 16);             // tensor_dim0[15:0]=64 in bits[31:16]
  g1[2] = (int)((ur & 0xFFFFu) << 16);          // tensor_dim0 hi=0 | tensor_dim1 lo
  g1[3] = (int)(((ur >> 16) & 0xFFFFu)          // tensor_dim1 hi
        |       ((uint32_t)D << 16));           // tile_dim0 = 64
  g1[4] = (int)(ur & 0xFFFFu);                  // tile_dim1 = rows ; tile_dim2 = 0
  g1[5] = D;                                    // tensor_dim0_stride lo = 64
  g1[6] = 0;                                    // stride hi / dim1_stride lo
  g1[7] = 0;
  i32x4 gz = {0, 0, 0, 0};
#if __clang_major__ >= 23
  i32x8 gz8 = {0, 0, 0, 0, 0, 0, 0, 0};
  __builtin_amdgcn_tensor_load_to_lds(g0, g1, gz, gz, gz8, 0);
#else
  __builtin_amdgcn_tensor_load_to_lds(g0, g1, gz, gz, 0);
#endif
}
#endif

__device__ __forceinline__ void acc4(const float4 v, float4& sum, float4& sq,
                                     float4& mx, float4& mn) {
  sum.x += v.x; sum.y += v.y; sum.z += v.z; sum.w += v.w;
  sq.x = fmaf(v.x, v.x, sq.x); sq.y = fmaf(v.y, v.y, sq.y);
  sq.z = fmaf(v.z, v.z, sq.z); sq.w = fmaf(v.w, v.w, sq.w);
  mx.x = fmaxf(mx.x, v.x); mx.y = fmaxf(mx.y, v.y);
  mx.z = fmaxf(mx.z, v.z); mx.w = fmaxf(mx.w, v.w);
  mn.x = fminf(mn.x, v.x); mn.y = fminf(mn.y, v.y);
  mn.z = fminf(mn.z, v.z); mn.w = fminf(mn.w, v.w);
}

__global__ __launch_bounds__(BLOCK) void seg_pool_kernel(
    const float* __restrict__ edges, const float* __restrict__ meta,
    const float* __restrict__ att, const float* __restrict__ cur,
    const int* __restrict__ eb, float* __restrict__ out, int E) {
  __shared__ float lds[NBUF * CH * D];  // 48 KB: TDM ring, buf0 reused for reduction
  __shared__ int   sb[2];

  const int g   = blockIdx.x;
  const int tid = threadIdx.x;

  // Segment bounds via binary search on the sorted edge_batch.
  if (tid < 2) {
    int key = g + tid, lo = 0, hi = E;
    while (lo < hi) {
      int m = (lo + hi) >> 1;
      if (eb[m] < key) lo = m + 1; else hi = m;
    }
    sb[tid] = lo;
  }

  const size_t base = (size_t)g * 640;

  // Pass-through rows issued up front so their vmem overlaps the TDM stream:
  // metadata -> [0,128), attended -> [384,512), current -> [512,640)
  {
    const float4* m4 = (const float4*)(meta + (size_t)g * 128);
    const float4* a4 = (const float4*)(att + (size_t)g * 128);
    const float4* c4 = (const float4*)(cur + (size_t)g * 128);
    float4* o4 = (float4*)(out + base);
    for (int j = tid; j < 96; j += BLOCK) {
      if (j < 32)      o4[j]              = m4[j];
      else if (j < 64) o4[96 + (j - 32)]  = a4[j - 32];
      else             o4[128 + (j - 64)] = c4[j - 64];
    }
  }

  __syncthreads();
  const int seg_lo = sb[0];
  const int nrows  = sb[1] - sb[0];

  const int r = tid >> 4;   // row-group 0..15
  const int s = tid & 15;   // float4 slot 0..15 (features 4s..4s+3)

  const float INF = __builtin_inff();
  float4 vsum = {0.f, 0.f, 0.f, 0.f};
  float4 vsq  = {0.f, 0.f, 0.f, 0.f};
  float4 vmx  = {-INF, -INF, -INF, -INF};
  float4 vmn  = { INF,  INF,  INF,  INF};

#if HAVE_TDM
  // Wave 0 drives the Tensor Data Mover; 3-deep ring of chunks in LDS.
  const bool   w0      = (__builtin_amdgcn_readfirstlane((int)(threadIdx.x >> 5)) == 0);
  const float* segbase = edges + (size_t)seg_lo * D;
  const int    nch     = (nrows + CH - 1) / CH;
  const int    npre    = nch < NBUF ? nch : NBUF;

  if (w0) {
    for (int i = 0; i < npre; ++i) {            // prime the pipeline
      int rn = nrows - i * CH; if (rn > CH) rn = CH;
      tdm_load_rows((uint32_t)(uintptr_t)&lds[i * CH * D],
                    segbase + (size_t)i * CH * D, rn);
    }
  }
  int nis = npre;                               // chunks issued so far (uniform)

  for (int c = 0; c < nch; ++c) {
    const int rem = nch - c;                    // chunks left, including c
    if (w0) {
      // Outstanding = min(NBUF, rem); TDM completes in order per wave, so
      // waiting tensorcnt <= outstanding-1 guarantees chunk c has landed.
      if (rem >= 3)      __builtin_amdgcn_s_wait_tensorcnt(2);
      else if (rem == 2) __builtin_amdgcn_s_wait_tensorcnt(1);
      else               __builtin_amdgcn_s_wait_tensorcnt(0);
    }
    __syncthreads();                            // publish chunk c to all waves

    int rows_c = nrows - c * CH; if (rows_c > CH) rows_c = CH;
    const float4* buf = (const float4*)&lds[(c % NBUF) * CH * D];
    for (int row = r; row < rows_c; row += 16) {
      float4 v = buf[row * 16 + s];             // ds_load_b128
      acc4(v, vsum, vsq, vmx, vmn);
    }
    __syncthreads();                            // compute(c) done -> buf[c%3] reusable

    if (w0 && nis < nch) {                      // refill ring; overlaps next computes
      int rn = nrows - nis * CH; if (rn > CH) rn = CH;
      tdm_load_rows((uint32_t)(uintptr_t)&lds[(nis % NBUF) * CH * D],
                    segbase + (size_t)nis * CH * D, rn);
    }
    if (nis < nch) ++nis;
  }
#else
  const float4* gp = (const float4*)(edges + (size_t)seg_lo * D);
  for (int row = r; row < nrows; row += 16) {
    float4 v = gp[row * 16 + s];
    acc4(v, vsum, vsq, vmx, vmn);
  }
  __syncthreads();
#endif

  // Cross row-group reduction through LDS: red[stat][16][64], 16 KB (reuses lds).
  {
    float* red = lds;
    const int f0 = s * 4;
    const float* ps = (const float*)&vsum;
    const float* pq = (const float*)&vsq;
    const float* px = (const float*)&vmx;
    const float* pn = (const float*)&vmn;
#pragma unroll
    for (int j = 0; j < 4; ++j) {
      red[0 * 1024 + r * 64 + f0 + j] = ps[j];
      red[1 * 1024 + r * 64 + f0 + j] = pq[j];
      red[2 * 1024 + r * 64 + f0 + j] = px[j];
      red[3 * 1024 + r * 64 + f0 + j] = pn[j];
    }
  }
  __syncthreads();

  if (tid < D) {
    const float* red = lds;
    float ssum = 0.f, ssq = 0.f, smx = -INF, smn = INF;
#pragma unroll
    for (int rr = 0; rr < 16; ++rr) {
      ssum += red[0 * 1024 + rr * 64 + tid];
      ssq  += red[1 * 1024 + rr * 64 + tid];
      smx = fmaxf(smx, red[2 * 1024 + rr * 64 + tid]);
      smn = fminf(smn, red[3 * 1024 + rr * 64 + tid]);
    }
    float mean = 0.f, var = 0.f, omx = 0.f, omn = 0.f;
    if (nrows > 0) {
      float inv = 1.0f / (float)nrows;
      mean = ssum * inv;
      var  = fmaf(-mean, mean, ssq * inv);   // E[x^2] - mean^2
      var  = fmaxf(var, 0.0f);
      omx  = smx;
      omn  = smn;
    }
    out[base + 128 + tid] = mean;
    out[base + 192 + tid] = var;
    out[base + 256 + tid] = omx;
    out[base + 320 + tid] = omn;
  }
}

extern "C" void kernel_launch(void* const* d_in, const int* in_sizes, int n_in,
                              void* d_out, int out_size, void* d_ws, size_t ws_size,
                              hipStream_t stream) {
  const float* edges = (const float*)d_in[0];
  const float* meta  = (const float*)d_in[1];
  const float* att   = (const float*)d_in[2];
  const float* cur   = (const float*)d_in[3];
  const int*   eb    = (const int*)d_in[4];
  const int E = in_sizes[0] / D;     // 2,000,000
  const int G = in_sizes[1] / 128;   // 8192
  seg_pool_kernel<<<dim3(G), dim3(BLOCK), 0, stream>>>(
      edges, meta, att, cur, eb, (float*)d_out, E);
}